// GAT_59854664237646
// MI455X (gfx1250) — compile-verified
//
#include <hip/hip_runtime.h>
#include <hip/hip_bf16.h>

typedef float v2f __attribute__((ext_vector_type(2)));
typedef float v8f __attribute__((ext_vector_type(8)));

#define NEG_SLOPE 0.2f

// ============================================================
// WMMA fp32 GEMM: C[M,NOUT] = X[M,K] @ W[K,NOUT]
// K, NOUT compile-time; one wave32 per 16 x (16*NT) output strip.
// A tile (X) is loaded once per k-step and reused across NT WMMAs.
// All global loads use compile-time immediate offsets (no per-iter
// 64-bit address arithmetic).
// VGPR layouts per CDNA5 ISA 7.12.2:
//   A 16x4 : lane L<16 holds (M=L, K=0..1), lane L>=16 holds (M=L-16, K=2..3)
//   B 4x16 : v0 = (K=0 | K=2), v1 = (K=1 | K=3), N = lane&15
//   C 16x16: VGPR r -> M = r + 8*(lane>=16), N = lane&15
// ============================================================
template<int K, int NOUT, int NT>
__global__ __launch_bounds__(256) void gemm_wmma_f32(
    const float* __restrict__ X, const float* __restrict__ W,
    float* __restrict__ C, int M)
{
  const int lane   = threadIdx.x & 31;
  const int wave   = threadIdx.x >> 5;
  constexpr int tilesN = NOUT / (16 * NT);
  const int tiles  = (M >> 4) * tilesN;
  const int tile   = blockIdx.x * 8 + wave;     // wave-uniform
  if (tile >= tiles) return;                     // uniform exit: EXEC stays ~0
  const int tM = tile / tilesN;
  const int tN = tile % tilesN;

  const int half = lane >> 4;      // 0 = lanes 0-15, 1 = lanes 16-31
  const int l15  = lane & 15;
  const float* xrow = X + (size_t)(tM * 16 + l15) * K + 2 * half;
  const float* wb   = W + (size_t)(2 * half) * NOUT + tN * (16 * NT) + l15;

  v8f acc[NT];
#pragma unroll
  for (int t = 0; t < NT; ++t) acc[t] = (v8f){0.f,0.f,0.f,0.f,0.f,0.f,0.f,0.f};

#pragma unroll
  for (int k = 0; k < K; k += 4) {
    v2f a;
    a.x = xrow[k + 0];           // one global_load_b64, imm offset
    a.y = xrow[k + 1];
#pragma unroll
    for (int t = 0; t < NT; ++t) {
      v2f b;
      b.x = wb[(size_t)(k + 0) * NOUT + 16 * t];   // imm-offset b32 loads
      b.y = wb[(size_t)(k + 1) * NOUT + 16 * t];
      // (neg_a, A, neg_b, B, c_mod, C, reuse_a, reuse_b)
      acc[t] = __builtin_amdgcn_wmma_f32_16x16x4_f32(false, a, false, b,
                                                     (short)0, acc[t],
                                                     false, false);
    }
  }

  const int mbase = tM * 16 + 8 * half;
  float* cp = C + (size_t)mbase * NOUT + tN * (16 * NT) + l15;
#pragma unroll
  for (int t = 0; t < NT; ++t)
#pragma unroll
    for (int r = 0; r < 8; ++r)
      cp[(size_t)r * NOUT + 16 * t] = acc[t][r];
}

// a[n] = dot(X[n,:], v)  — one wave per row, shuffle reduction (wave32)
__global__ __launch_bounds__(256) void rowdot(
    const float* __restrict__ X, const float* __restrict__ v,
    float* __restrict__ out, int N, int K)
{
  const int gw   = (int)((blockIdx.x * (size_t)blockDim.x + threadIdx.x) >> 5);
  const int lane = threadIdx.x & 31;
  if (gw >= N) return;
  const float* row = X + (size_t)gw * K;
  float s = 0.f;
  for (int k = lane; k < K; k += 32) s += row[k] * v[k];
#pragma unroll
  for (int off = 16; off > 0; off >>= 1) s += __shfl_down(s, off, 32);
  if (lane == 0) out[gw] = s;
}

// v[j] = dot(W[j,:], att)  — fold W_dst @ att_dst into a single vector
__global__ void foldvec(const float* __restrict__ W, const float* __restrict__ att,
                        float* __restrict__ v, int rows, int cols)
{
  const int j = blockIdx.x * blockDim.x + threadIdx.x;
  if (j >= rows) return;
  const float* wr = W + (size_t)j * cols;
  float s = 0.f;
  for (int k = 0; k < cols; ++k) s += wr[k] * att[k];
  v[j] = s;
}

__global__ void init_seg(float* __restrict__ m, float* __restrict__ denom, int N)
{
  const int i = blockIdx.x * blockDim.x + threadIdx.x;
  if (i < N) { m[i] = -__builtin_huge_valf(); denom[i] = 0.f; }
}

__global__ void init_bias(float* __restrict__ out, const float* __restrict__ b,
                          int total, int C)
{
  const int i = blockIdx.x * blockDim.x + threadIdx.x;
  if (i < total) out[i] = b[i % C];
}

__global__ void relu_ip(float* __restrict__ x, int n)
{
  const int i = blockIdx.x * blockDim.x + threadIdx.x;
  if (i < n) x[i] = fmaxf(x[i], 0.f);
}

// float atomic max via sign-split integer atomics (lowers to
// global_atomic_max_i32 / global_atomic_min_u32; exact for IEEE floats, no NaN)
__device__ __forceinline__ void atomicMaxF(float* addr, float val)
{
  if (val >= 0.f) atomicMax((int*)addr, __float_as_int(val));
  else            atomicMin((unsigned int*)addr, __float_as_uint(val));
}

__global__ __launch_bounds__(256) void edge_logits(
    const int* __restrict__ src, const int* __restrict__ dst,
    const float* __restrict__ asrc, const float* __restrict__ adst,
    float* __restrict__ e, float* __restrict__ m, int E)
{
  const int i = blockIdx.x * blockDim.x + threadIdx.x;
  if (i >= E) return;
  const int d = dst[i];
  float v = asrc[src[i]] + adst[d];
  v = (v > 0.f) ? v : NEG_SLOPE * v;
  e[i] = v;
  atomicMaxF(&m[d], v);
}

__global__ __launch_bounds__(256) void edge_exp(
    const int* __restrict__ dst, const float* __restrict__ m,
    float* __restrict__ e, float* __restrict__ denom, int E)
{
  const int i = blockIdx.x * blockDim.x + threadIdx.x;
  if (i >= E) return;
  const int d = dst[i];
  const float ee = __expf(e[i] - m[d]);
  e[i] = ee;
  atomicAdd(&denom[d], ee);
}

// out[dst] += alpha * h[src], one wave per edge, float4/float2 gathers,
// C/32 f32 atomics per lane.
template<int C>
__global__ __launch_bounds__(256) void edge_scatter(
    const int* __restrict__ src, const int* __restrict__ dst,
    const float* __restrict__ ee, const float* __restrict__ denom,
    const float* __restrict__ h, float* __restrict__ out, int E)
{
  const int wv   = (int)((blockIdx.x * (size_t)blockDim.x + threadIdx.x) >> 5);
  const int lane = threadIdx.x & 31;
  if (wv >= E) return;
  const int s = src[wv];
  const int d = dst[wv];
  const float alpha = ee[wv] / denom[d];
  constexpr int P = C / 32;  // 4 (C=128) or 2 (C=64)
  const float* hp = h   + (size_t)s * C + lane * P;
  float*       op = out + (size_t)d * C + lane * P;
  float tmp[P];
  if constexpr (P == 4) {
    const float4 v = *(const float4*)hp;
    tmp[0] = v.x; tmp[1] = v.y; tmp[2] = v.z; tmp[3] = v.w;
  } else {
    const float2 v = *(const float2*)hp;
    tmp[0] = v.x; tmp[1] = v.y;
  }
#pragma unroll
  for (int i = 0; i < P; ++i) atomicAdd(op + i, alpha * tmp[i]);
}

static inline int cdiv(long a, long b) { return (int)((a + b - 1) / b); }

extern "C" void kernel_launch(void* const* d_in, const int* in_sizes, int n_in,
                              void* d_out, int out_size, void* d_ws, size_t ws_size,
                              hipStream_t stream)
{
  const float* x       = (const float*)d_in[0];
  const int*   ei      = (const int*)d_in[1];
  const float* Wsrc1   = (const float*)d_in[2];
  const float* Wdst1   = (const float*)d_in[3];
  const float* attsrc1 = (const float*)d_in[4];
  const float* attdst1 = (const float*)d_in[5];
  const float* b1      = (const float*)d_in[6];
  const float* Wsrc2   = (const float*)d_in[7];
  const float* Wdst2   = (const float*)d_in[8];
  const float* attsrc2 = (const float*)d_in[9];
  const float* attdst2 = (const float*)d_in[10];
  const float* b2      = (const float*)d_in[11];

  const int HID = in_sizes[4];            // 128
  const int OUT = in_sizes[9];            // 64
  const int FIN = in_sizes[2] / HID;      // 128
  const int N   = in_sizes[0] / FIN;      // 100000 (divisible by 16)
  const int E   = in_sizes[1] / 2;        // 1.6M
  const int* src = ei;
  const int* dst = ei + E;

  // workspace carve-out (floats)
  float* ws    = (float*)d_ws;
  float* h1    = ws;  ws += (size_t)N * HID;   // x @ Wsrc1
  float* o1    = ws;  ws += (size_t)N * HID;   // layer-1 output (relu'd in place)
  float* h2    = ws;  ws += (size_t)N * OUT;   // relu(o1) @ Wsrc2
  float* asrc  = ws;  ws += N;
  float* adst  = ws;  ws += N;
  float* mbuf  = ws;  ws += N;
  float* dbuf  = ws;  ws += N;
  float* ebuf  = ws;  ws += E;
  float* vfold = ws;  ws += HID;

  // ---------------- layer 1 ----------------
  {
    const int tiles = (N / 16) * (HID / 64);  // NT=4 -> 16x64 strips
    gemm_wmma_f32<128, 128, 4><<<cdiv(tiles, 8), 256, 0, stream>>>(x, Wsrc1, h1, N);
    rowdot<<<cdiv((long)N * 32, 256), 256, 0, stream>>>(h1, attsrc1, asrc, N, HID);
    foldvec<<<cdiv(FIN, 128), 128, 0, stream>>>(Wdst1, attdst1, vfold, FIN, HID);
    rowdot<<<cdiv((long)N * 32, 256), 256, 0, stream>>>(x, vfold, adst, N, FIN);
    init_seg<<<cdiv(N, 256), 256, 0, stream>>>(mbuf, dbuf, N);
    init_bias<<<cdiv((long)N * HID, 256), 256, 0, stream>>>(o1, b1, N * HID, HID);
    edge_logits<<<cdiv(E, 256), 256, 0, stream>>>(src, dst, asrc, adst, ebuf, mbuf, E);
    edge_exp<<<cdiv(E, 256), 256, 0, stream>>>(dst, mbuf, ebuf, dbuf, E);
    edge_scatter<128><<<cdiv((long)E * 32, 256), 256, 0, stream>>>(
        src, dst, ebuf, dbuf, h1, o1, E);
    relu_ip<<<cdiv((long)N * HID, 256), 256, 0, stream>>>(o1, N * HID);
  }

  // ---------------- layer 2 ----------------
  {
    float* out = (float*)d_out;
    const int tiles = (N / 16) * (OUT / 64);  // NT=4 -> one 16x64 strip per row-block
    gemm_wmma_f32<128, 64, 4><<<cdiv(tiles, 8), 256, 0, stream>>>(o1, Wsrc2, h2, N);
    rowdot<<<cdiv((long)N * 32, 256), 256, 0, stream>>>(h2, attsrc2, asrc, N, OUT);
    foldvec<<<cdiv(HID, 128), 128, 0, stream>>>(Wdst2, attdst2, vfold, HID, OUT);
    rowdot<<<cdiv((long)N * 32, 256), 256, 0, stream>>>(o1, vfold, adst, N, HID);
    init_seg<<<cdiv(N, 256), 256, 0, stream>>>(mbuf, dbuf, N);
    init_bias<<<cdiv((long)N * OUT, 256), 256, 0, stream>>>(out, b2, N * OUT, OUT);
    edge_logits<<<cdiv(E, 256), 256, 0, stream>>>(src, dst, asrc, adst, ebuf, mbuf, E);
    edge_exp<<<cdiv(E, 256), 256, 0, stream>>>(dst, mbuf, ebuf, dbuf, E);
    edge_scatter<64><<<cdiv((long)E * 32, 256), 256, 0, stream>>>(
        src, dst, ebuf, dbuf, h2, out, E);
  }
}